// ButterflyNetwork_79233556676748
// MI455X (gfx1250) — compile-verified
//
#include <hip/hip_runtime.h>
#include <math.h>

// ---------------- problem constants (from reference) ----------------
#define COLB   16
#define IN_W   1024
#define OUT_W  512
#define DEPTH  8
#define IN_L   4
#define OUT_L  4
#define ACT    8
#define BLOCKS 64
#define GROW   (BLOCKS * ACT)            // 512
#define TOTAL  (IN_W + DEPTH * GROW)     // 5120
#define BATCH  4096
#define WROWS  (BLOCKS * COLB)           // 1024

typedef float v2f __attribute__((ext_vector_type(2)));
typedef float v8f __attribute__((ext_vector_type(8)));

// ------------------------------------------------------------------
// Phase 0: data[0:1024] = scales[:,None] * input   (float4 vectorized)
// ------------------------------------------------------------------
__global__ __launch_bounds__(256)
void bfly_scale_input(const float4* __restrict__ in,
                      const float*  __restrict__ scales,
                      float4* __restrict__ data)
{
    int i = blockIdx.x * 256 + threadIdx.x;      // over 1024 * (4096/4) float4
    int row = i >> 10;                           // 1024 float4 per row
    float s = scales[row];
    float4 v = in[i];
    v.x *= s; v.y *= s; v.z *= s; v.w *= s;
    data[i] = v;
}

// ------------------------------------------------------------------
// Phase 1: build the 16x16 block rotation matrices (one-time, tiny).
// 4 butterfly layers (strides 1,2,4,8) compose into a dense 16x16.
// One thread builds one COLUMN of one matrix: R[:,c] = L3 L2 L1 L0 e_c.
// rmats layout: [d][blk][io][16][16] row-major, io=0 -> Rin, io=1 -> Rout.
// Precise sinf/cosf kept intentionally: runs once, feeds 8 chained GEMMs.
// ------------------------------------------------------------------
__global__ __launch_bounds__(256)
void bfly_build_rot(const float* __restrict__ angles,   // [DEPTH][8][512]
                    float* __restrict__ rmats)          // [DEPTH][64][2][256]
{
    int t   = blockIdx.x * 256 + threadIdx.x;   // 16384 threads total
    int c   = t & 15;
    int blk = (t >> 4) & 63;
    int io  = (t >> 10) & 1;
    int d   = t >> 11;

    float col[16];
#pragma unroll
    for (int m = 0; m < 16; ++m) col[m] = (m == c) ? 1.0f : 0.0f;

    const float* angBase = angles + (size_t)d * 8 * 512 + (size_t)io * 4 * 512;
#pragma unroll
    for (int l = 0; l < 4; ++l) {
        const float* a = angBase + l * 512 + blk * 8;
        const int s = 1 << l;                   // strides 1,2,4,8 (in AND out)
#pragma unroll
        for (int p = 0; p < 8; ++p) {
            int gl = p / s, j = p % s;          // constants after unroll
            int lo = gl * 2 * s + j, hi = lo + s;
            float cc = cosf(a[p]);
            float ss = sinf(a[p]);
            float L = col[lo], H = col[hi];
            col[lo] =  cc * L + ss * H;
            col[hi] = -ss * L + cc * H;
        }
    }
    float* out = rmats + ((((size_t)d * 64 + blk) * 2 + io) * 256);
#pragma unroll
    for (int m = 0; m < 16; ++m) out[m * 16 + c] = col[m];
}

// ------------------------------------------------------------------
// Phase 2: one module.
//   y = Rin_blk * x(gathered)            4x V_WMMA_F32_16X16X4_F32
//   y[0:8] = 0.5*(p + sqrt(p*p+1)), p = y+bias     (C-layout, lanes 0-15)
//   z = Rout_blk * y                     4x WMMA (via LDS C->B relayout)
//   data[idx] = z ; actOut rows = activations
//
// grid = (64 blocks, 8), block = 128 (4 wave32); 32 waves per block, each
// wave owns 8 of the 256 16-column strips, strip stride 32 -> a constant
// 2048-byte column advance per unrolled iteration, folded into the 24-bit
// instruction immediate offset (no per-iteration address VALU).
//
// WMMA operand maps (ISA 7.12.2, wave32):
//   A(m,k):  lane = m + 16*(k/2),  vgpr = k%2
//   B(k,n):  lane = n + 16*(k/2),  vgpr = k%2
//   C/D(m,n):lane = n + 16*(m/8),  vgpr = m%8
// ------------------------------------------------------------------
template <bool DOSCATTER>
__global__ __launch_bounds__(128)
void bfly_module(float* __restrict__ data,
                 const float* __restrict__ rmatsDepth,  // [64][2][256]
                 const float* __restrict__ bias,        // [512]
                 const int*   __restrict__ idx,         // [1024]
                 float* __restrict__ actOut)            // act rows, BATCH stride
{
    const int blk  = blockIdx.x;         // 0..63
    const int wgy  = blockIdx.y;         // 0..7
    const int tid  = threadIdx.x;
    const int wave = tid >> 5;           // 0..3
    const int lane = tid & 31;
    const int nl   = lane & 15;          // n (or m for A)
    const int hf   = lane >> 4;          // lane-half

    __shared__ float lds[4][16 * 20];    // stride 20 -> conflict-free halves
    float* tile = lds[wave];

    // ---- per-block constants (hoisted out of the strip loop) ----
    const float* Ri = rmatsDepth + (size_t)blk * 512;
    const float* Ro = Ri + 256;

    float ain[8], aout[8];
#pragma unroll
    for (int q = 0; q < 8; ++q) {        // q = 2*k0 + j
        int k = 4 * (q >> 1) + 2 * hf + (q & 1);
        ain[q] = Ri[nl * 16 + k];
        if (DOSCATTER) aout[q] = Ro[nl * 16 + k];
    }

    const int* idxB = idx + blk * 16;
    const int s0 = wgy * 4 + wave;                       // first strip (0..31)
    const unsigned colByte = (unsigned)s0 * 64u + (unsigned)nl * 4u;

    // per-lane base pointers; all loop movement goes into immediate offsets
    const char* gptr[8];
#pragma unroll
    for (int q = 0; q < 8; ++q) {
        int r = 4 * (q >> 1) + 2 * hf + (q & 1);         // B-operand row
        gptr[q] = (const char*)data + (unsigned)idxB[r] * (BATCH * 4u) + colByte;
    }
    char* sptr[8];
    if (DOSCATTER) {
#pragma unroll
        for (int v = 0; v < 8; ++v)                      // D row m = v + 8*hf
            sptr[v] = (char*)data + (unsigned)idxB[v + 8 * hf] * (BATCH * 4u) + colByte;
    }
    float* aptr = actOut + (size_t)blk * ACT * BATCH + (size_t)s0 * 16 + nl;

    float bv[8];                         // block-uniform -> scalar loads
#pragma unroll
    for (int v = 0; v < 8; ++v) bv[v] = bias[blk * ACT + v];

    // ---- fully unrolled strip loop: 8 iterations, +2048 B per step ----
#pragma unroll
    for (int it = 0; it < 8; ++it) {
        const int ofs = it * 2048;                       // byte immediate

        // gather x in B layout
        v2f b[4];
#pragma unroll
        for (int k0 = 0; k0 < 4; ++k0) {
            b[k0].x = *(const float*)(gptr[2 * k0 + 0] + ofs);
            b[k0].y = *(const float*)(gptr[2 * k0 + 1] + ofs);
        }

        // y = Rin * x
        v8f acc = {};
#pragma unroll
        for (int k0 = 0; k0 < 4; ++k0) {
            v2f a; a.x = ain[2 * k0]; a.y = ain[2 * k0 + 1];
            acc = __builtin_amdgcn_wmma_f32_16x16x4_f32(
                false, a, false, b[k0], (short)0, acc, false, false);
        }

        // activation on rows 0..7 (C-layout: vgpr v, lanes 0..15)
        // hardware v_sqrt_f32 (~1 ulp) instead of libm fixup sequence
        float act[8];
#pragma unroll
        for (int v = 0; v < 8; ++v) {
            float pre = acc[v] + bv[v];
            float sq  = __builtin_amdgcn_sqrtf(__builtin_fmaf(pre, pre, 1.0f));
            float a2  = __builtin_fmaf(0.5f, sq, 0.5f * pre);
            act[v] = a2;
            acc[v] = (hf == 0) ? a2 : acc[v];
        }
        if (hf == 0) {
#pragma unroll
            for (int v = 0; v < 8; ++v)
                aptr[v * BATCH + it * 512] = act[v];     // immediate offsets
        }

        if (DOSCATTER) {
            // C-layout -> LDS tile (row stride 20 floats: halves hit
            // disjoint bank sets for both store and load patterns)
#pragma unroll
            for (int v = 0; v < 8; ++v)
                tile[(v + 8 * hf) * 20 + nl] = acc[v];

            asm volatile("s_wait_dscnt 0" ::: "memory");

            // reload y in B layout
            v2f b2[4];
#pragma unroll
            for (int k0 = 0; k0 < 4; ++k0) {
                int r0 = 4 * k0 + 2 * hf;
                b2[k0].x = tile[(r0 + 0) * 20 + nl];
                b2[k0].y = tile[(r0 + 1) * 20 + nl];
            }

            // z = Rout * y
            v8f z = {};
#pragma unroll
            for (int k0 = 0; k0 < 4; ++k0) {
                v2f a; a.x = aout[2 * k0]; a.y = aout[2 * k0 + 1];
                z = __builtin_amdgcn_wmma_f32_16x16x4_f32(
                    false, a, false, b2[k0], (short)0, z, false, false);
            }

            // scatter z back to data[idx]
#pragma unroll
            for (int v = 0; v < 8; ++v)
                *(float*)(sptr[v] + ofs) = z[v];
        }
    }
}

// ------------------------------------------------------------------
// launcher
// inputs: 0 input_data (1024x4096 f32), 1 scales (1024 f32),
//         2 angles (8x8x512 f32), 3 biases (8x512 f32), 4 indices (8x1024 i32)
// output: 512x4096 f32 (== activations of module 7)
// ws: [0, 80MB) data buffer, then [80MB, +1MB) rotation matrices
// ------------------------------------------------------------------
extern "C" void kernel_launch(void* const* d_in, const int* in_sizes, int n_in,
                              void* d_out, int out_size, void* d_ws, size_t ws_size,
                              hipStream_t stream)
{
    (void)in_sizes; (void)n_in; (void)out_size; (void)ws_size;

    const float* input  = (const float*)d_in[0];
    const float* scales = (const float*)d_in[1];
    const float* angles = (const float*)d_in[2];
    const float* biases = (const float*)d_in[3];
    const int*   indices= (const int*)  d_in[4];
    float* out = (float*)d_out;

    const size_t DATA_BYTES = (size_t)TOTAL * BATCH * sizeof(float); // 80 MB
    float* data  = (float*)d_ws;
    float* rmats = (float*)((char*)d_ws + DATA_BYTES);               // 1 MB

    // phase 0: scale inputs into data[0:1024]
    {
        int nvec = IN_W * BATCH / 4;
        bfly_scale_input<<<nvec / 256, 256, 0, stream>>>(
            (const float4*)input, scales, (float4*)data);
    }

    // phase 1: build all 1024 rotation matrices
    bfly_build_rot<<<(DEPTH * BLOCKS * 2 * 16) / 256, 256, 0, stream>>>(
        angles, rmats);

    // phase 2: eight sequential modules
    for (int i = 0; i < DEPTH; ++i) {
        const float* rd = rmats + (size_t)i * BLOCKS * 2 * 256;
        const float* bd = biases + (size_t)i * GROW;
        const int*   id = indices + (size_t)i * WROWS;
        if (i < DEPTH - 1) {
            float* actOut = data + (size_t)(IN_W + GROW * i) * BATCH;
            bfly_module<true><<<dim3(BLOCKS, 8), 128, 0, stream>>>(
                data, rd, bd, id, actOut);
        } else {
            // last module: its activations ARE the output; out-rotations unused
            bfly_module<false><<<dim3(BLOCKS, 8), 128, 0, stream>>>(
                data, rd, bd, id, out);
        }
    }
}